// MixtralSparseMoeBlockEP_47029891891652
// MI455X (gfx1250) — compile-verified
//
#include <hip/hip_runtime.h>
#include <math.h>

typedef float v2f __attribute__((ext_vector_type(2)));
typedef float v8f __attribute__((ext_vector_type(8)));

#define H_DIM 1024
#define F_DIM 2048
#define E_NUM 8
#define T_NUM 2048
#define TM 32
#define TN 64
#define KC 32
#define LDK 36  // pad: row stride 144B (16B-aligned for async B128, conflict-free frags)

static __device__ __forceinline__ v8f wmma_f32(v2f a, v2f b, v8f c) {
  // D = A(16x4) * B(4x16) + C, all fp32, RNE accumulate (matches fp32 reference)
  return __builtin_amdgcn_wmma_f32_16x16x4_f32(false, a, false, b, (short)0, c,
                                               false, false);
}

// CDNA5 async copy: global -> LDS, tracked by ASYNCcnt (no VGPR staging).
// LDS byte offset = low 32 bits of the flat shared address (ISA aperture rule).
static __device__ __forceinline__ void async_copy_b128(const float* gaddr,
                                                       const float* lds_ptr) {
  unsigned l = (unsigned)(uintptr_t)lds_ptr;
  asm volatile("global_load_async_to_lds_b128 %0, %1, off"
               :: "v"(l), "v"(gaddr)
               : "memory");
}

static __device__ __forceinline__ void wait_async0() {
  asm volatile("s_wait_asynccnt 0" ::: "memory");
}

// ---------------------------------------------------------------------------
// Kernel 0: reset per-expert atomic counters (ws is never re-poisoned)
// ---------------------------------------------------------------------------
__global__ void moe_zero_counts(int* __restrict__ counts) {
  if (threadIdx.x < E_NUM) counts[threadIdx.x] = 0;
}

// ---------------------------------------------------------------------------
// Kernel 1: router. One wave32 per token: logits = x . gate_w^T, softmax,
// top-2 (first-index-on-tie, like jax.lax.top_k), renormalize. Writes logits
// into d_out and appends assignment ids (2t+k) into per-expert lists.
// ---------------------------------------------------------------------------
__global__ __launch_bounds__(256) void moe_router(
    const float* __restrict__ x, const float* __restrict__ gate_w,
    float* __restrict__ logits_out, int* __restrict__ counts,
    int* __restrict__ idxbuf, float* __restrict__ cw) {
  const int wid = threadIdx.x >> 5;
  const int lane = threadIdx.x & 31;
  const int token = blockIdx.x * 8 + wid;
  if (token >= T_NUM) return;

  float acc[E_NUM];
#pragma unroll
  for (int e = 0; e < E_NUM; ++e) acc[e] = 0.0f;

  const float* xr = x + (size_t)token * H_DIM;
#pragma unroll 4
  for (int i = 0; i < H_DIM / 32; ++i) {
    const int h = i * 32 + lane;
    const float xv = xr[h];
#pragma unroll
    for (int e = 0; e < E_NUM; ++e) acc[e] += xv * gate_w[e * H_DIM + h];
  }
#pragma unroll
  for (int e = 0; e < E_NUM; ++e) {
#pragma unroll
    for (int off = 16; off > 0; off >>= 1) acc[e] += __shfl_xor(acc[e], off, 32);
  }

  // softmax (f32, max-subtracted) — redundantly on all lanes, lane 0 writes
  float m = acc[0];
#pragma unroll
  for (int e = 1; e < E_NUM; ++e) m = fmaxf(m, acc[e]);
  float p[E_NUM];
#pragma unroll
  for (int e = 0; e < E_NUM; ++e) p[e] = expf(acc[e] - m);

  int e0 = 0;
  float b0 = p[0];
#pragma unroll
  for (int e = 1; e < E_NUM; ++e)
    if (p[e] > b0) { b0 = p[e]; e0 = e; }
  int e1 = -1;
  float b1 = -1.0f;
#pragma unroll
  for (int e = 0; e < E_NUM; ++e)
    if (e != e0 && p[e] > b1) { b1 = p[e]; e1 = e; }

  const float wsum = b0 + b1;  // softmax denominator cancels in the ratio
  if (lane == 0) {
#pragma unroll
    for (int e = 0; e < E_NUM; ++e) logits_out[token * E_NUM + e] = acc[e];
    const int s0 = atomicAdd(&counts[e0], 1);
    idxbuf[e0 * T_NUM + s0] = token * 2;
    cw[token * 2] = b0 / wsum;
    const int s1 = atomicAdd(&counts[e1], 1);
    idxbuf[e1 * T_NUM + s1] = token * 2 + 1;
    cw[token * 2 + 1] = b1 / wsum;
  }
}

// ---------------------------------------------------------------------------
// Kernel 2: gathered expert GEMM1 + SwiGLU.
//   mid[aid, f] = silu(x[t].w1[e,f,:]) * (x[t].w3[e,f,:])
// Block = 256 thr (8 waves), tile 32 tokens x 64 F, K chunk = 32.
// Double-buffered LDS fed by global_load_async_to_lds_b128: copy of chunk
// N+1 overlaps the WMMAs of chunk N; one barrier + one asynccnt wait/chunk.
// ---------------------------------------------------------------------------
__global__ __launch_bounds__(256) void moe_gemm1(
    const float* __restrict__ x, const float* __restrict__ w1,
    const float* __restrict__ w3, const int* __restrict__ counts,
    const int* __restrict__ idxbuf, float* __restrict__ mid) {
  __shared__ float sx[2][TM * LDK];
  __shared__ float sw1[2][TN * LDK];
  __shared__ float sw3[2][TN * LDK];
  __shared__ int s_aid[TM];
  __shared__ int s_tok[TM];

  const int e = blockIdx.z;
  const int cnt = counts[e];
  const int slot0 = blockIdx.y * TM;
  if (slot0 >= cnt) return;
  const int n0 = blockIdx.x * TN;
  const int tid = threadIdx.x;

  if (tid < TM) {
    const int slot = slot0 + tid;
    const int aid = (slot < cnt) ? idxbuf[e * T_NUM + slot] : -1;
    s_aid[tid] = aid;
    s_tok[tid] = (aid >= 0) ? (aid >> 1) : 0;  // safe dummy row
  }
  __syncthreads();

  // per-thread load coordinates
  const int xr = tid >> 3;            // x tile: 32 rows x 8 float4
  const int xc = (tid & 7) * 4;
  const int my_tok = s_tok[xr];
  const size_t wbase0 = ((size_t)(e * F_DIM + n0 + (tid >> 3))) * H_DIM + xc;
  const size_t wbase1 =
      ((size_t)(e * F_DIM + n0 + ((tid + 256) >> 3))) * H_DIM + ((tid & 7) * 4);

  const int lane = tid & 31;
  const int wid = tid >> 5;
  const int mt = wid & 1;   // M tile (0/1)
  const int np = wid >> 1;  // N tile (0..3), 16 cols each
  const int half = lane >> 4;
  const int l15 = lane & 15;

  v8f ah = {}, ag = {};

  const int NCH = H_DIM / KC;
  // prologue: async-fill buffer 0 with chunk 0
  {
    async_copy_b128(x + (size_t)my_tok * H_DIM + xc, &sx[0][xr * LDK + xc]);
    async_copy_b128(w1 + wbase0, &sw1[0][(tid >> 3) * LDK + xc]);
    async_copy_b128(w3 + wbase0, &sw3[0][(tid >> 3) * LDK + xc]);
    async_copy_b128(w1 + wbase1, &sw1[0][((tid + 256) >> 3) * LDK + xc]);
    async_copy_b128(w3 + wbase1, &sw3[0][((tid + 256) >> 3) * LDK + xc]);
  }

#pragma unroll 1
  for (int chunk = 0; chunk < NCH; ++chunk) {
    wait_async0();   // my copies for this chunk are done (in-order)
    __syncthreads(); // everyone's copies visible; everyone done with prev buf
    if (chunk + 1 < NCH) {
      const int nb = (chunk + 1) & 1;
      const int k1 = (chunk + 1) * KC;
      async_copy_b128(x + (size_t)my_tok * H_DIM + k1 + xc,
                      &sx[nb][xr * LDK + xc]);
      async_copy_b128(w1 + wbase0 + k1, &sw1[nb][(tid >> 3) * LDK + xc]);
      async_copy_b128(w3 + wbase0 + k1, &sw3[nb][(tid >> 3) * LDK + xc]);
      async_copy_b128(w1 + wbase1 + k1, &sw1[nb][((tid + 256) >> 3) * LDK + xc]);
      async_copy_b128(w3 + wbase1 + k1, &sw3[nb][((tid + 256) >> 3) * LDK + xc]);
    }
    const int buf = chunk & 1;
    const float* ap = &sx[buf][(mt * 16 + l15) * LDK];
    const float* p1 = &sw1[buf][(np * 16 + l15) * LDK];
    const float* p3 = &sw3[buf][(np * 16 + l15) * LDK];
#pragma unroll
    for (int kk = 0; kk < KC; kk += 4) {
      const int kb = kk + half * 2;
      v2f a;  a.x = ap[kb];  a.y = ap[kb + 1];
      v2f b1; b1.x = p1[kb]; b1.y = p1[kb + 1];
      v2f b3; b3.x = p3[kb]; b3.y = p3[kb + 1];
      ah = wmma_f32(a, b1, ah);
      ag = wmma_f32(a, b3, ag);
    }
  }

#pragma unroll
  for (int r = 0; r < 8; ++r) {
    const int mrow = mt * 16 + r + half * 8;
    const int aid = s_aid[mrow];
    if (aid < 0) continue;
    const int ng = n0 + np * 16 + l15;
    const float h = ah[r], g = ag[r];
    mid[(size_t)aid * F_DIM + ng] = (h / (1.0f + expf(-h))) * g;
  }
}

// ---------------------------------------------------------------------------
// Kernel 3: gathered expert GEMM2, combine weight folded into epilogue.
//   eo[aid, h] = cw[aid] * (mid[aid,:] . w2[e,h,:])
// Same async double-buffered structure; padded rows zero-filled once.
// ---------------------------------------------------------------------------
__global__ __launch_bounds__(256) void moe_gemm2(
    const float* __restrict__ mid, const float* __restrict__ w2,
    const int* __restrict__ counts, const int* __restrict__ idxbuf,
    const float* __restrict__ cw, float* __restrict__ eo) {
  __shared__ float sa[2][TM * LDK];
  __shared__ float sb[2][TN * LDK];
  __shared__ int s_aid[TM];

  const int e = blockIdx.z;
  const int cnt = counts[e];
  const int slot0 = blockIdx.y * TM;
  if (slot0 >= cnt) return;
  const int n0 = blockIdx.x * TN;  // over H
  const int tid = threadIdx.x;

  if (tid < TM) {
    const int slot = slot0 + tid;
    s_aid[tid] = (slot < cnt) ? idxbuf[e * T_NUM + slot] : -1;
  }
  __syncthreads();

  const int ar = tid >> 3;
  const int ac = (tid & 7) * 4;
  const int my_aid = s_aid[ar];
  const size_t bbase0 = ((size_t)(e * H_DIM + n0 + (tid >> 3))) * F_DIM + ac;
  const size_t bbase1 =
      ((size_t)(e * H_DIM + n0 + ((tid + 256) >> 3))) * F_DIM + ((tid & 7) * 4);

  // padded rows are never async-loaded: zero both buffers once
  if (my_aid < 0) {
    float* d0 = &sa[0][ar * LDK + ac];
    float* d1 = &sa[1][ar * LDK + ac];
#pragma unroll
    for (int q = 0; q < 4; ++q) { d0[q] = 0.0f; d1[q] = 0.0f; }
  }

  const int lane = tid & 31;
  const int wid = tid >> 5;
  const int mt = wid & 1;
  const int np = wid >> 1;
  const int half = lane >> 4;
  const int l15 = lane & 15;

  v8f acc = {};

  const int NCH = F_DIM / KC;
  {  // prologue: chunk 0 into buffer 0
    if (my_aid >= 0)
      async_copy_b128(mid + (size_t)my_aid * F_DIM + ac, &sa[0][ar * LDK + ac]);
    async_copy_b128(w2 + bbase0, &sb[0][(tid >> 3) * LDK + ac]);
    async_copy_b128(w2 + bbase1, &sb[0][((tid + 256) >> 3) * LDK + ac]);
  }

#pragma unroll 1
  for (int chunk = 0; chunk < NCH; ++chunk) {
    wait_async0();
    __syncthreads();
    if (chunk + 1 < NCH) {
      const int nb = (chunk + 1) & 1;
      const int k1 = (chunk + 1) * KC;
      if (my_aid >= 0)
        async_copy_b128(mid + (size_t)my_aid * F_DIM + k1 + ac,
                        &sa[nb][ar * LDK + ac]);
      async_copy_b128(w2 + bbase0 + k1, &sb[nb][(tid >> 3) * LDK + ac]);
      async_copy_b128(w2 + bbase1 + k1, &sb[nb][((tid + 256) >> 3) * LDK + ac]);
    }
    const int buf = chunk & 1;
    const float* ap = &sa[buf][(mt * 16 + l15) * LDK];
    const float* bp = &sb[buf][(np * 16 + l15) * LDK];
#pragma unroll
    for (int kk = 0; kk < KC; kk += 4) {
      const int kb = kk + half * 2;
      v2f av; av.x = ap[kb]; av.y = ap[kb + 1];
      v2f bv; bv.x = bp[kb]; bv.y = bp[kb + 1];
      acc = wmma_f32(av, bv, acc);
    }
  }

#pragma unroll
  for (int r = 0; r < 8; ++r) {
    const int mrow = mt * 16 + r + half * 8;
    const int aid = s_aid[mrow];
    if (aid < 0) continue;
    const int ng = n0 + np * 16 + l15;
    eo[(size_t)aid * H_DIM + ng] = cw[aid] * acc[r];
  }
}

// ---------------------------------------------------------------------------
// Kernel 4: deterministic combine: out[t] = eo[2t] + eo[2t+1]
// ---------------------------------------------------------------------------
__global__ void moe_combine(const float* __restrict__ eo,
                            float* __restrict__ out) {
  const int i = blockIdx.x * blockDim.x + threadIdx.x;  // t*H + h
  const int h = i & (H_DIM - 1);
  const int t = i >> 10;
  const size_t a0 = ((size_t)t * 2) * H_DIM + h;
  out[i] = eo[a0] + eo[a0 + H_DIM];
}

// ---------------------------------------------------------------------------
extern "C" void kernel_launch(void* const* d_in, const int* in_sizes, int n_in,
                              void* d_out, int out_size, void* d_ws,
                              size_t ws_size, hipStream_t stream) {
  (void)in_sizes; (void)n_in; (void)out_size; (void)ws_size;
  const float* x      = (const float*)d_in[0];  // [T, H]
  const float* gate_w = (const float*)d_in[1];  // [E, H]
  const float* w1     = (const float*)d_in[2];  // [E, F, H]
  const float* w2     = (const float*)d_in[3];  // [E, H, F]
  const float* w3     = (const float*)d_in[4];  // [E, F, H]

  float* out    = (float*)d_out;                // [T, H]
  float* logits = out + (size_t)T_NUM * H_DIM;  // [T, E]

  char* ws = (char*)d_ws;
  const size_t OFF_IDX = 256;                                      // E*T ints
  const size_t OFF_CW  = OFF_IDX + (size_t)E_NUM * T_NUM * 4;      // 2T floats
  const size_t OFF_MID = OFF_CW + (size_t)2 * T_NUM * 4;           // 2T*F
  const size_t OFF_EO  = OFF_MID + (size_t)2 * T_NUM * F_DIM * 4;  // 2T*H
  int*   counts = (int*)(ws);
  int*   idxbuf = (int*)(ws + OFF_IDX);
  float* cw     = (float*)(ws + OFF_CW);
  float* mid    = (float*)(ws + OFF_MID);
  float* eo     = (float*)(ws + OFF_EO);

  moe_zero_counts<<<1, 32, 0, stream>>>(counts);
  moe_router<<<T_NUM / 8, 256, 0, stream>>>(x, gate_w, logits, counts, idxbuf,
                                            cw);
  dim3 g1(F_DIM / TN, T_NUM / TM, E_NUM);
  moe_gemm1<<<g1, 256, 0, stream>>>(x, w1, w3, counts, idxbuf, mid);
  dim3 g2(H_DIM / TN, T_NUM / TM, E_NUM);
  moe_gemm2<<<g2, 256, 0, stream>>>(mid, w2, counts, idxbuf, cw, eo);
  moe_combine<<<(T_NUM * H_DIM) / 256, 256, 0, stream>>>(eo, out);
}